// LCNN_20847771255049
// MI455X (gfx1250) — compile-verified
//
#include <hip/hip_runtime.h>
#include <hip/hip_bf16.h>

#define LOG2S  0.6931f
#define BN_EPS 1e-5f
#define LN_EPS 1e-5f

typedef __attribute__((ext_vector_type(2))) float v2f;
typedef __attribute__((ext_vector_type(8))) float v8f;
typedef int v4i __attribute__((vector_size(16)));   // async b128 payload type

typedef __attribute__((address_space(1))) v4i g_v4i;  // global (device) int4
typedef __attribute__((address_space(3))) v4i l_v4i;  // LDS int4

static constexpr int P    = 6;
static constexpr int K    = 19;
static constexpr int PK   = P * K;    // 114 indices per site
static constexpr int FOUT = 19;       // conv output features
static constexpr int SF   = 25;       // sitewise features

#if __has_builtin(__builtin_amdgcn_global_load_async_to_lds_b128) && \
    __has_builtin(__builtin_amdgcn_s_wait_asynccnt)
#define HAVE_ASYNC_LDS 1
#else
#define HAVE_ASYNC_LDS 0
#endif

// ---------------------------------------------------------------------------
// Lattice conv block with the permutation sum folded into the gather and the
// eval-mode BatchNorm folded into an affine epilogue:
//   S[n,k,f]  = sum_p node[nbr[n,p,k], f]
//   out[n,o]  = alpha_o * (S[n,:] . W[:,o]) + beta_o
//   alpha_o   = g_o * rsqrt(rv_o + eps)
//   beta_o    = P * (alpha_o * (b_o - rm_o) + be_o)
// GEMM runs on V_WMMA_F32_16X16X4_F32: M=16 sites/wave, K zero-padded in LDS
// to a multiple of 4 (so no B-value masking is needed, only index clamps),
// N=19 covered by two 16-wide tiles (tile1 columns >= 19 computed but never
// stored). The 32-site nbr_idx chunk is contiguous and staged into LDS with
// GLOBAL_LOAD_ASYNC_TO_LDS_B128 (ASYNCcnt) when the builtin is available.
// ---------------------------------------------------------------------------
template<int FIN, int KF, int KF4, int STRIDE>
__global__ __launch_bounds__(64)
void lcnn_block_kernel(const float* __restrict__ node,
                       const int*   __restrict__ nbr,
                       const float* __restrict__ W,
                       const float* __restrict__ b,
                       const float* __restrict__ g,
                       const float* __restrict__ be,
                       const float* __restrict__ rm,
                       const float* __restrict__ rv,
                       float* __restrict__ out,
                       int nSites)
{
    __shared__ float S[32 * STRIDE];
    __shared__ int   sIdx[32 * PK];           // 14.25 KB staged indices
    const int tid  = threadIdx.x;
    const int base = blockIdx.x * 32;

    // ---- stage the contiguous 32-site index chunk into LDS ----------------
    const int idxCount = 32 * PK;             // 3648 dwords
    const long gIdxBase = (long)base * PK;
    const bool fullBlk = (gIdxBase + idxCount) <= (long)nSites * PK;
#if HAVE_ASYNC_LDS
    if (fullBlk) {
        // 3648 dwords = 912 b128 transfers, 64 lanes cooperate
        for (int i = tid; i < idxCount / 4; i += 64) {
            __builtin_amdgcn_global_load_async_to_lds_b128(
                (g_v4i*)(nbr + gIdxBase + 4 * i),
                (l_v4i*)(&sIdx[4 * i]),
                0, 0);
        }
    } else
#endif
    {
        for (int i = tid; i < idxCount; i += 64) {
            const long gi = gIdxBase + i;
            sIdx[i] = (gi < (long)nSites * PK) ? nbr[gi] : 0;
        }
    }

    // Zero S (covers the K padding tail of every row) while the async copy
    // is in flight.
    for (int i = tid; i < 32 * STRIDE; i += 64) S[i] = 0.f;

#if HAVE_ASYNC_LDS
    if (fullBlk) __builtin_amdgcn_s_wait_asynccnt(0);
#endif
    __syncthreads();

    // ---- gather + fold the permutation sum --------------------------------
    // Consecutive threads walk f fastest, so the 6 gathered rows are read as
    // coalesced FIN-float bursts (sources are 1.2-7.6 MB: fully L2-resident),
    // and the FIN-thread f-group broadcasts a single LDS index word.
    for (int e = tid; e < 32 * KF; e += 64) {
        const int s   = e / KF;
        const int rem = e - s * KF;
        const int k   = rem / FIN;
        const int f   = rem - k * FIN;
        const int site = base + s;
        float acc = 0.f;
        if (site < nSites) {
            const int* row = sIdx + s * PK + k;
            #pragma unroll
            for (int p = 0; p < P; ++p) {
                const int j = row[p * K];
                acc += node[(size_t)j * FIN + f];
            }
        }
        S[s * STRIDE + rem] = acc;
    }
    __syncthreads();

    // ---- per-wave [16 x KF] @ [KF x 19] on the fp32 WMMA pipe -------------
    const int wave = tid >> 5;
    const int lane = tid & 31;
    const int m    = lane & 15;
    const int hi   = lane >> 4;
    const int kb   = hi << 1;          // K sub-offset for this half-wave
    const int n0   = lane & 15;        // tile0 column
    const int n1c  = (n0 < FOUT - 16) ? n0 + 16 : 0;  // tile1 column (clamped)

    const float* Srow = S + (wave * 16 + m) * STRIDE;
    v8f acc0 = {0.f,0.f,0.f,0.f,0.f,0.f,0.f,0.f};
    v8f acc1 = {0.f,0.f,0.f,0.f,0.f,0.f,0.f,0.f};

    for (int kk = 0; kk < KF4; kk += 4) {
        const int kr = kk + kb;
        v2f a;
        a.x = Srow[kr];
        a.y = Srow[kr + 1];
        // A rows >= KF are zero in LDS, so B needs only an in-bounds clamp
        // (no value masking) -> no cndmask chain, EXEC untouched.
        const int kr0 = min(kr,     KF - 1);
        const int kr1 = min(kr + 1, KF - 1);
        v2f b0, b1;
        b0.x = W[kr0 * FOUT + n0];
        b0.y = W[kr1 * FOUT + n0];
        b1.x = W[kr0 * FOUT + n1c];
        b1.y = W[kr1 * FOUT + n1c];
        acc0 = __builtin_amdgcn_wmma_f32_16x16x4_f32(false, a, false, b0,
                                                     (short)0, acc0, false, false);
        acc1 = __builtin_amdgcn_wmma_f32_16x16x4_f32(false, a, false, b1,
                                                     (short)0, acc1, false, false);
    }

    // ---- folded-BN epilogue (per-lane output column n0 / n1) --------------
    const float inv0   = rsqrtf(rv[n0] + BN_EPS);
    const float alpha0 = g[n0] * inv0;
    const float beta0  = (float)P * (alpha0 * (b[n0] - rm[n0]) + be[n0]);
    const float inv1   = rsqrtf(rv[n1c] + BN_EPS);
    const float alpha1 = g[n1c] * inv1;
    const float beta1  = (float)P * (alpha1 * (b[n1c] - rm[n1c]) + be[n1c]);
    const bool  hasT1  = (n0 + 16) < FOUT;

    #pragma unroll
    for (int r = 0; r < 8; ++r) {
        const int srow = base + wave * 16 + r + 8 * hi;  // D: M = r + 8*(lane/16)
        if (srow < nSites) {
            out[(size_t)srow * FOUT + n0] = alpha0 * acc0[r] + beta0;
            if (hasT1)
                out[(size_t)srow * FOUT + n0 + 16] = alpha1 * acc1[r] + beta1;
        }
    }
}

// ---------------------------------------------------------------------------
// Head: per site 19->25 linear, LayerNorm, shifted softplus, 25->25 linear,
// then a deterministic wave->block reduction of the site sum into d_ws.
// ---------------------------------------------------------------------------
__global__ __launch_bounds__(256)
void head_kernel(const float* __restrict__ h,
                 const float* __restrict__ Wc, const float* __restrict__ bc,
                 const float* __restrict__ lng, const float* __restrict__ lnb,
                 const float* __restrict__ Wl,
                 float* __restrict__ partials, int nSites)
{
    __shared__ float sWc[FOUT * SF];
    __shared__ float sWl[SF * SF];
    __shared__ float sbc[SF];
    __shared__ float swacc[8][SF];
    const int tid = threadIdx.x;

    for (int i = tid; i < FOUT * SF; i += 256) sWc[i] = Wc[i];
    for (int i = tid; i < SF * SF;   i += 256) sWl[i] = Wl[i];
    if (tid < SF) sbc[tid] = bc[tid];
    __syncthreads();

    const int  n   = blockIdx.x * 256 + tid;
    const bool act = n < nSites;
    float v[SF];
    if (act) {
        float hb[FOUT];
        #pragma unroll
        for (int i = 0; i < FOUT; ++i) hb[i] = h[(size_t)n * FOUT + i];
        #pragma unroll
        for (int o = 0; o < SF; ++o) {
            float acc = sbc[o];
            #pragma unroll
            for (int i = 0; i < FOUT; ++i) acc += hb[i] * sWc[i * SF + o];
            v[o] = acc;
        }
        // LayerNorm over the 25 features
        float mu = 0.f;
        #pragma unroll
        for (int o = 0; o < SF; ++o) mu += v[o];
        mu *= (1.f / SF);
        float var = 0.f;
        #pragma unroll
        for (int o = 0; o < SF; ++o) { const float d = v[o] - mu; var += d * d; }
        var *= (1.f / SF);
        const float rinv = rsqrtf(var + LN_EPS);
        #pragma unroll
        for (int o = 0; o < SF; ++o) {
            const float t = lng[o] * (v[o] - mu) * rinv + lnb[o];
            // numerically stable shifted softplus
            v[o] = fmaxf(t, 0.f) + log1pf(expf(-fabsf(t))) - LOG2S;
        }
    } else {
        #pragma unroll
        for (int o = 0; o < SF; ++o) v[o] = 0.f;
    }

    const int wid = tid >> 5, lane = tid & 31;
    #pragma unroll
    for (int q = 0; q < SF; ++q) {
        float wq = 0.f;
        if (act) {
            #pragma unroll
            for (int i = 0; i < SF; ++i) wq += v[i] * sWl[i * SF + q];
        }
        #pragma unroll
        for (int off = 16; off > 0; off >>= 1) wq += __shfl_xor(wq, off, 32);
        if (lane == 0) swacc[wid][q] = wq;
    }
    __syncthreads();
    if (tid < SF) {
        float s = 0.f;
        #pragma unroll
        for (int w = 0; w < 8; ++w) s += swacc[w][tid];
        partials[(size_t)blockIdx.x * SF + tid] = s;
    }
}

// ---------------------------------------------------------------------------
// Final: deterministic reduction of block partials, mean over sites, +bl,
// dot with Wf, +bf -> scalar output.
// ---------------------------------------------------------------------------
__global__ void finish_kernel(const float* __restrict__ partials, int nParts,
                              const float* __restrict__ bl,
                              const float* __restrict__ Wf,
                              const float* __restrict__ bf,
                              float* __restrict__ out, int nSites)
{
    const int lane = threadIdx.x;
    float val = 0.f;
    if (lane < SF) {
        float s = 0.f;
        for (int i = 0; i < nParts; ++i) s += partials[(size_t)i * SF + lane];
        const float gq = s / (float)nSites + bl[lane];
        val = gq * Wf[lane];
    }
    for (int off = 16; off > 0; off >>= 1) val += __shfl_xor(val, off, 32);
    if (lane == 0) out[0] = val + bf[0];
}

extern "C" void kernel_launch(void* const* d_in, const int* in_sizes, int n_in,
                              void* d_out, int out_size, void* d_ws, size_t ws_size,
                              hipStream_t stream)
{
    const float* x   = (const float*)d_in[0];
    const int*   nbr = (const int*)  d_in[1];
    const float* W1  = (const float*)d_in[2];
    const float* b1  = (const float*)d_in[3];
    const float* g1  = (const float*)d_in[4];
    const float* be1 = (const float*)d_in[5];
    const float* rm1 = (const float*)d_in[6];
    const float* rv1 = (const float*)d_in[7];
    const float* W2  = (const float*)d_in[8];
    const float* b2  = (const float*)d_in[9];
    const float* g2  = (const float*)d_in[10];
    const float* be2 = (const float*)d_in[11];
    const float* rm2 = (const float*)d_in[12];
    const float* rv2 = (const float*)d_in[13];
    const float* Wc  = (const float*)d_in[14];
    const float* bc  = (const float*)d_in[15];
    const float* lng = (const float*)d_in[16];
    const float* lnb = (const float*)d_in[17];
    const float* Wl  = (const float*)d_in[18];
    const float* bl  = (const float*)d_in[19];
    const float* Wf  = (const float*)d_in[20];
    const float* bf  = (const float*)d_in[21];

    const int nSites = in_sizes[0] / 3;   // x is [N, 3]

    // Workspace layout: h1 [N*19] | h2 [N*19] | partials [nHeadBlocks*25]
    float* h1 = (float*)d_ws;
    float* h2 = h1 + (size_t)nSites * FOUT;
    float* partials = h2 + (size_t)nSites * FOUT;

    const int nConvBlocks = (nSites + 31) / 32;
    // Block 1: F_in=3,  KF=57,  pad4=60,  LDS row stride 61 (odd -> bank-friendly)
    lcnn_block_kernel<3, 57, 60, 61><<<nConvBlocks, 64, 0, stream>>>(
        x, nbr, W1, b1, g1, be1, rm1, rv1, h1, nSites);
    // Block 2: F_in=19, KF=361, pad4=364, LDS row stride 365 (46.7+14.25 KB LDS)
    lcnn_block_kernel<19, 361, 364, 365><<<nConvBlocks, 64, 0, stream>>>(
        h1, nbr, W2, b2, g2, be2, rm2, rv2, h2, nSites);

    const int nHeadBlocks = (nSites + 255) / 256;
    head_kernel<<<nHeadBlocks, 256, 0, stream>>>(h2, Wc, bc, lng, lnb, Wl,
                                                 partials, nSites);
    finish_kernel<<<1, 32, 0, stream>>>(partials, nHeadBlocks, bl, Wf, bf,
                                        (float*)d_out, nSites);
}